// HippocampalModule_24043226923365
// MI455X (gfx1250) — compile-verified
//
#include <hip/hip_runtime.h>

#define DIM   256
#define HID   1024
#define CAP   1000000
#define TOPK  32
#define EPSF  1e-8f

#define CHUNK 8192
#define NBLK  ((CAP + CHUNK - 1) / CHUNK)   // 123
#define NCAND (NBLK * TOPK)                 // 3936

#define NEG_INF (-__builtin_inff())

typedef __attribute__((ext_vector_type(2))) float v2f;
typedef __attribute__((ext_vector_type(4))) float v4f;
typedef __attribute__((ext_vector_type(8))) float v8f;

// ---------------------------------------------------------------------------
// f32 WMMA 16x16x4 wrapper (codegen-confirmed round 1: lowers to
// v_wmma_f32_16x16x4_f32).  D = A(16x4) * B(4x16) + C(16x16), all f32.
// Lane layout (CDNA5 ISA §7.12.2):
//   A: lane l<16 holds A[m=l][k={0,1}], lane l>=16 holds A[m=l-16][k={2,3}]
//   B: lane l<16 holds B[k={0,1}][n=l], lane l>=16 holds B[k={2,3}][n=l-16]
//   C/D: reg r, lanes0-15 -> (m=r, n=lane); lanes16-31 -> (m=r+8, n=lane-16)
// ---------------------------------------------------------------------------
__device__ __forceinline__ v8f wmma_f32_16x16x4(v2f a, v2f b, v8f c) {
#if __has_builtin(__builtin_amdgcn_wmma_f32_16x16x4_f32)
  return __builtin_amdgcn_wmma_f32_16x16x4_f32(
      /*neg_a=*/false, a, /*neg_b=*/false, b,
      /*c_mod=*/(short)0, c, /*reuse_a=*/false, /*reuse_b=*/false);
#else
  c[0] = fmaf(a.x, b.x, c[0]);
  return c;
#endif
}

// ---------------------------------------------------------------------------
// Kernel 1: encode cue with MLP, write enc (256) and 1/max(||enc||,eps).
// One block of 1024 threads; every matvec column read is coalesced.
// ---------------------------------------------------------------------------
__global__ void k_encode(const float* __restrict__ cue,
                         const float* __restrict__ w_e1, const float* __restrict__ b_e1,
                         const float* __restrict__ w_e2, const float* __restrict__ b_e2,
                         float* __restrict__ enc_out, float* __restrict__ inv_cue_out) {
  __shared__ float cueS[DIM];
  __shared__ float hS[HID];
  __shared__ float red[DIM];
  const int t = threadIdx.x;  // blockDim == 1024
  if (t < DIM) cueS[t] = cue[t];
  __syncthreads();

  float acc = b_e1[t];
#pragma unroll 8
  for (int i = 0; i < DIM; ++i) acc = fmaf(cueS[i], w_e1[i * HID + t], acc);
  hS[t] = fmaxf(acc, 0.0f);
  __syncthreads();

  float e = 0.0f;
  if (t < DIM) {
    e = b_e2[t];
#pragma unroll 8
    for (int i = 0; i < HID; ++i) e = fmaf(hS[i], w_e2[i * DIM + t], e);
    enc_out[t] = e;
    red[t] = e * e;
  }
  __syncthreads();
  for (int s = DIM / 2; s > 0; s >>= 1) {
    if (t < s) red[t] += red[t + s];
    __syncthreads();
  }
  if (t == 0) *inv_cue_out = 1.0f / fmaxf(__builtin_sqrtf(red[0]), EPSF);
}

// ---------------------------------------------------------------------------
// Kernel 2: the HBM-bound scan (~1.02 GB @ 23.3 TB/s ~= 44 us lower bound).
// One row per wave per iteration; each lane holds 8 enc values in registers
// and streams 32B of the row (fully coalesced 1KB per wave).  Fused dot +
// sum-of-squares, wave32 xor-shuffle reduction.  The memories stream has
// zero reuse -> non-temporal loads so the 1 GB stream doesn't evict the
// 4 MB sims array (which the top-k pass re-reads) from the 192 MB L2.
// ---------------------------------------------------------------------------
__global__ void k_scan(const float* __restrict__ mem,
                       const unsigned char* __restrict__ mask,
                       const float* __restrict__ enc,
                       const float* __restrict__ inv_cue_ptr,
                       float* __restrict__ sims) {
  const int lane = threadIdx.x & 31;
  const int wib = threadIdx.x >> 5;
  const int wpb = blockDim.x >> 5;
  const long long gwave = (long long)blockIdx.x * wpb + wib;
  const long long nwaves = (long long)gridDim.x * wpb;
  const float inv_cue = *inv_cue_ptr;

  const v4f e0 = ((const v4f*)enc)[lane * 2 + 0];
  const v4f e1 = ((const v4f*)enc)[lane * 2 + 1];

  for (long long r = gwave; r < CAP; r += nwaves) {
    const v4f* row = (const v4f*)(mem + (size_t)r * DIM);
    if (r + nwaves < CAP)
      __builtin_prefetch(mem + (size_t)(r + nwaves) * DIM + lane * 8, 0, 0);
    const v4f a0 = __builtin_nontemporal_load(&row[lane * 2 + 0]);
    const v4f a1 = __builtin_nontemporal_load(&row[lane * 2 + 1]);

    float dot = 0.0f, ss = 0.0f;
    dot = fmaf(a0.x, e0.x, dot); ss = fmaf(a0.x, a0.x, ss);
    dot = fmaf(a0.y, e0.y, dot); ss = fmaf(a0.y, a0.y, ss);
    dot = fmaf(a0.z, e0.z, dot); ss = fmaf(a0.z, a0.z, ss);
    dot = fmaf(a0.w, e0.w, dot); ss = fmaf(a0.w, a0.w, ss);
    dot = fmaf(a1.x, e1.x, dot); ss = fmaf(a1.x, a1.x, ss);
    dot = fmaf(a1.y, e1.y, dot); ss = fmaf(a1.y, a1.y, ss);
    dot = fmaf(a1.z, e1.z, dot); ss = fmaf(a1.z, a1.z, ss);
    dot = fmaf(a1.w, e1.w, dot); ss = fmaf(a1.w, a1.w, ss);

#pragma unroll
    for (int m = 16; m >= 1; m >>= 1) {
      dot += __shfl_xor(dot, m, 32);
      ss  += __shfl_xor(ss,  m, 32);
    }
    if (lane == 0) {
      float sim = dot / fmaxf(__builtin_sqrtf(ss), EPSF) * inv_cue;
      unsigned char mk = __builtin_nontemporal_load(&mask[r]);
      sims[r] = mk ? sim : NEG_INF;   // regular store: keep sims resident in L2
    }
  }
}

// ---------------------------------------------------------------------------
// Kernel 3: per-block top-32 over an 8192-row chunk of sims (LDS argmax x32).
// sims values are consumed exactly once -> non-temporal reads.
// ---------------------------------------------------------------------------
__global__ void k_localtopk(const float* __restrict__ sims,
                            float* __restrict__ candV, int* __restrict__ candI) {
  __shared__ float s[CHUNK];
  __shared__ float rv[256];
  __shared__ int   rp[256];
  const int t = threadIdx.x;  // blockDim == 256
  const long long base = (long long)blockIdx.x * CHUNK;
  for (int i = t; i < CHUNK; i += 256) {
    long long r = base + i;
    s[i] = (r < CAP) ? __builtin_nontemporal_load(&sims[r]) : NEG_INF;
  }
  __syncthreads();
  for (int k = 0; k < TOPK; ++k) {
    float bv = s[t];
    int bp = t;
    for (int i = t + 256; i < CHUNK; i += 256) {
      float v = s[i];
      if (v > bv) { bv = v; bp = i; }
    }
    rv[t] = bv; rp[t] = bp;
    __syncthreads();
    for (int sN = 128; sN > 0; sN >>= 1) {
      if (t < sN && rv[t + sN] > rv[t]) { rv[t] = rv[t + sN]; rp[t] = rp[t + sN]; }
      __syncthreads();
    }
    if (t == 0) {
      candV[blockIdx.x * TOPK + k] = rv[0];
      candI[blockIdx.x * TOPK + k] = (int)(base + rp[0]);
      s[rp[0]] = NEG_INF;
    }
    __syncthreads();
  }
}

// ---------------------------------------------------------------------------
// Kernel 4: merge 3936 candidates to final top-32; emit sims + indices.
// ---------------------------------------------------------------------------
__global__ void k_finaltopk(const float* __restrict__ candV, const int* __restrict__ candI,
                            float* __restrict__ out_sims, float* __restrict__ out_idx_f,
                            int* __restrict__ top_idx) {
  __shared__ float s[NCAND];
  __shared__ int   si[NCAND];
  __shared__ float rv[256];
  __shared__ int   rp[256];
  const int t = threadIdx.x;  // blockDim == 256
  for (int i = t; i < NCAND; i += 256) { s[i] = candV[i]; si[i] = candI[i]; }
  __syncthreads();
  for (int k = 0; k < TOPK; ++k) {
    float bv = s[t];
    int bp = t;
    for (int i = t + 256; i < NCAND; i += 256) {
      float v = s[i];
      if (v > bv) { bv = v; bp = i; }
    }
    rv[t] = bv; rp[t] = bp;
    __syncthreads();
    for (int sN = 128; sN > 0; sN >>= 1) {
      if (t < sN && rv[t + sN] > rv[t]) { rv[t] = rv[t + sN]; rp[t] = rp[t + sN]; }
      __syncthreads();
    }
    if (t == 0) {
      int idx = si[rp[0]];
      out_sims[k] = rv[0];
      out_idx_f[k] = (float)idx;
      top_idx[k] = idx;
      s[rp[0]] = NEG_INF;
    }
    __syncthreads();
  }
}

// ---------------------------------------------------------------------------
// Kernel 5: decode MLP on the 32 retrieved rows with f32 WMMA 16x16x4.
// One block, 8 waves.  Hidden layer processed in 8 chunks of 128 columns:
// layer1 -> LDS (relu), barrier, layer2 accumulates into per-wave v8f regs.
// LDS strides padded (260 / 132) to avoid 64-bank conflicts on column reads.
// EXEC is all-ones around every WMMA (uniform control flow, 256 threads).
// ---------------------------------------------------------------------------
#define XS 260
#define HS 132

__global__ void k_decode(const float* __restrict__ mem, const int* __restrict__ top_idx,
                         const float* __restrict__ w_d1, const float* __restrict__ b_d1,
                         const float* __restrict__ w_d2, const float* __restrict__ b_d2,
                         float* __restrict__ out) {
  __shared__ float Xs[32 * XS];
  __shared__ float Hs[32 * HS];
  __shared__ int   idxS[TOPK];

  const int t = threadIdx.x;        // blockDim == 256 (8 waves)
  const int wave = t >> 5;
  const int lane = t & 31;
  const int half = lane >> 4;       // 0: lanes 0-15, 1: lanes 16-31
  const int l16 = lane & 15;

  if (t < TOPK) idxS[t] = top_idx[t];
  __syncthreads();

  // Gather the 32 retrieved memory rows into LDS.
#pragma unroll
  for (int i = 0; i < 32; ++i) {
    int p = t + i * 256;            // p in [0, 8192)
    int row = p >> 8, col = p & 255;
    Xs[row * XS + col] = mem[(size_t)idxS[row] * DIM + col];
  }
  __syncthreads();

  // Output accumulators: 32 output tiles (2 M x 16 N), 4 per wave.
  v8f accs[4];
  accs[0] = (v8f){}; accs[1] = (v8f){}; accs[2] = (v8f){}; accs[3] = (v8f){};

  for (int kc = 0; kc < HID / 128; ++kc) {
    // ---- Layer 1: H[:, kc*128 .. kc*128+127] = relu(X @ W1 + b1) ----
#pragma unroll
    for (int i = 0; i < 2; ++i) {
      const int tile = wave + 8 * i;       // 0..15
      const int m0 = (tile & 1) * 16;
      const int n0g = kc * 128 + (tile >> 1) * 16;
      v8f c = {};
      for (int kk = 0; kk < DIM / 4; ++kk) {
        const int k0 = kk * 4 + 2 * half;
        v2f a, b;
        a.x = Xs[(m0 + l16) * XS + k0 + 0];
        a.y = Xs[(m0 + l16) * XS + k0 + 1];
        b.x = w_d1[(size_t)(k0 + 0) * HID + n0g + l16];
        b.y = w_d1[(size_t)(k0 + 1) * HID + n0g + l16];
        c = wmma_f32_16x16x4(a, b, c);
      }
      const int nl = (tile >> 1) * 16 + l16;          // 0..127 in chunk
      const float bias = b_d1[kc * 128 + nl];
#pragma unroll
      for (int r = 0; r < 8; ++r) {
        const int m = m0 + r + 8 * half;
        Hs[m * HS + nl] = fmaxf(c[r] + bias, 0.0f);
      }
    }
    __syncthreads();

    // ---- Layer 2: out += H_chunk @ W2[kc*128 .. , :] ----
#pragma unroll
    for (int j = 0; j < 4; ++j) {
      const int tile = wave + 8 * j;       // 0..31
      const int m0 = (tile & 1) * 16;
      const int n0 = (tile >> 1) * 16;
      v8f c = accs[j];
      for (int kk = 0; kk < 128 / 4; ++kk) {
        const int k0l = kk * 4 + 2 * half;
        v2f a, b;
        a.x = Hs[(m0 + l16) * HS + k0l + 0];
        a.y = Hs[(m0 + l16) * HS + k0l + 1];
        b.x = w_d2[(size_t)(kc * 128 + k0l + 0) * DIM + n0 + l16];
        b.y = w_d2[(size_t)(kc * 128 + k0l + 1) * DIM + n0 + l16];
        c = wmma_f32_16x16x4(a, b, c);
      }
      accs[j] = c;
    }
    __syncthreads();
  }

  // Write decoded output + final bias.
#pragma unroll
  for (int j = 0; j < 4; ++j) {
    const int tile = wave + 8 * j;
    const int m0 = (tile & 1) * 16;
    const int n0 = (tile >> 1) * 16;
    const float bias = b_d2[n0 + l16];
#pragma unroll
    for (int r = 0; r < 8; ++r) {
      const int m = m0 + r + 8 * half;
      out[m * DIM + n0 + l16] = accs[j][r] + bias;
    }
  }
}

// ---------------------------------------------------------------------------
extern "C" void kernel_launch(void* const* d_in, const int* in_sizes, int n_in,
                              void* d_out, int out_size, void* d_ws, size_t ws_size,
                              hipStream_t stream) {
  const float* cue  = (const float*)d_in[0];
  const float* mem  = (const float*)d_in[1];
  const unsigned char* mask = (const unsigned char*)d_in[2];  // jnp.bool_: 1 byte/elem
  const float* w_e1 = (const float*)d_in[3];
  const float* b_e1 = (const float*)d_in[4];
  const float* w_e2 = (const float*)d_in[5];
  const float* b_e2 = (const float*)d_in[6];
  const float* w_d1 = (const float*)d_in[7];
  const float* b_d1 = (const float*)d_in[8];
  const float* w_d2 = (const float*)d_in[9];
  const float* b_d2 = (const float*)d_in[10];

  float* out = (float*)d_out;                 // [decoded 8192 | sims 32 | idx 32]
  float* wsf = (float*)d_ws;
  float* enc     = wsf;                       // 256
  float* inv_cue = wsf + 256;                 // 1
  float* candV   = wsf + 512;                 // 3936
  int*   candI   = (int*)(wsf + 4608);        // 3936
  int*   topIdx  = (int*)(wsf + 8704);        // 32
  float* sims    = wsf + 9216;                // 1,000,000

  k_encode<<<1, 1024, 0, stream>>>(cue, w_e1, b_e1, w_e2, b_e2, enc, inv_cue);
  k_scan<<<4096, 256, 0, stream>>>(mem, mask, enc, inv_cue, sims);
  k_localtopk<<<NBLK, 256, 0, stream>>>(sims, candV, candI);
  k_finaltopk<<<1, 256, 0, stream>>>(candV, candI, out + 8192, out + 8192 + TOPK, topIdx);
  k_decode<<<1, 256, 0, stream>>>(mem, topIdx, w_d1, b_d1, w_d2, b_d2, out);
}